// GraphAttentionLayer_36249523978474
// MI455X (gfx1250) — compile-verified
//
#include <hip/hip_runtime.h>
#include <hip/hip_bf16.h>

typedef __attribute__((ext_vector_type(2))) float v2f;
typedef __attribute__((ext_vector_type(8))) float v8f;

#define LRELU_ALPHA 0.2f
#define EXP_CLAMP 1000000.0f
#define DENOM_EPS 1e-10f
#define IN_F 128
#define OUT_F 64

// ---------------------------------------------------------------------------
// Kernel 0: zero-init denom[N] and out[N*64] (accumulated in-place via atomics)
// ---------------------------------------------------------------------------
__global__ void gat_zero_init(float* __restrict__ out, float* __restrict__ denom,
                              int n_out, int n_denom) {
    int i = blockIdx.x * blockDim.x + threadIdx.x;
    if (i < n_out) out[i] = 0.0f;
    if (i < n_denom) denom[i] = 0.0f;
}

// ---------------------------------------------------------------------------
// Kernel 1: Wh = h @ W  via V_WMMA_F32_16X16X4_F32
// One wave (32 lanes) computes one 16x16 tile of Wh. K=128 -> 32 WMMA steps.
// A 16x4 f32 layout: lanes 0-15 -> M=lane, VGPR0/1 = K=0/1; lanes 16-31 -> K=2/3
// B 4x16 f32 layout (symmetric): lane%16 = N, half selects K pair
// C/D 16x16 f32:  VGPR r -> M = r + 8*(lane/16), N = lane%16
// ---------------------------------------------------------------------------
__global__ void gat_wh_gemm(const float* __restrict__ h, const float* __restrict__ W,
                            float* __restrict__ Wh, int N) {
    const int lane = threadIdx.x & 31;
    const int waveInBlock = threadIdx.x >> 5;
    const int waveId = blockIdx.x * (blockDim.x >> 5) + waveInBlock;

    const int tilesN = OUT_F / 16;           // 4 tiles along N
    const int mTile = waveId / tilesN;
    const int nTile = waveId % tilesN;
    const int m0 = mTile * 16;
    if (m0 >= N) return;
    const int n0 = nTile * 16;

    const int nr = lane & 15;   // M index for A-frag, N index for B/D-frags
    const int hl = lane >> 4;   // half: selects K pair {0,1} vs {2,3}

    v8f acc = {};
    const float* hrow = h + (size_t)(m0 + nr) * IN_F;

    for (int k0 = 0; k0 < IN_F; k0 += 4) {
        const int kk = k0 + 2 * hl;
        // A fragment: A[M=nr][kk], A[M=nr][kk+1]  (contiguous -> b64 load)
        v2f a;
        a.x = hrow[kk + 0];
        a.y = hrow[kk + 1];
        // B fragment: B[kk][n0+nr], B[kk+1][n0+nr]
        v2f b;
        b.x = W[(size_t)(kk + 0) * OUT_F + n0 + nr];
        b.y = W[(size_t)(kk + 1) * OUT_F + n0 + nr];
        acc = __builtin_amdgcn_wmma_f32_16x16x4_f32(
            /*neg_a=*/false, a, /*neg_b=*/false, b,
            /*c_mod=*/(short)0, acc, /*reuse_a=*/false, /*reuse_b=*/false);
    }

#pragma unroll
    for (int r = 0; r < 8; ++r) {
        const int m = r + 8 * hl;
        Wh[(size_t)(m0 + m) * OUT_F + n0 + nr] = acc[r];
    }
}

// ---------------------------------------------------------------------------
// Kernel 2: s[i] = Wh[i,:] . a1 ; t[i] = Wh[i,:] . a2
// ---------------------------------------------------------------------------
__global__ void gat_st(const float* __restrict__ Wh, const float* __restrict__ a_vec,
                       float* __restrict__ s, float* __restrict__ t, int N) {
    int i = blockIdx.x * blockDim.x + threadIdx.x;
    if (i >= N) return;
    const float* r = Wh + (size_t)i * OUT_F;
    float ss = 0.0f, tt = 0.0f;
#pragma unroll
    for (int j = 0; j < OUT_F; ++j) {
        float v = r[j];
        ss += v * a_vec[j];
        tt += v * a_vec[OUT_F + j];
    }
    s[i] = ss;
    t[i] = tt;
}

// ---------------------------------------------------------------------------
// Kernel 3: per-edge attention logits -> x_exp, atomic segment-sum into denom
// ---------------------------------------------------------------------------
__global__ void gat_edge_exp(const long long* __restrict__ adj,
                             const float* __restrict__ s, const float* __restrict__ t,
                             float* __restrict__ x_exp, float* __restrict__ denom, int E) {
    int e = blockIdx.x * blockDim.x + threadIdx.x;
    if (e >= E) return;
    long long dst = adj[2 * (size_t)e + 0];
    long long src = adj[2 * (size_t)e + 1];
    float ev = s[dst] + t[src];
    ev = ev > 0.0f ? ev : LRELU_ALPHA * ev;
    float x = __expf(ev);
    x = fminf(x, EXP_CLAMP);           // exp >= 0, so lower clip is implicit
    x_exp[e] = x;
    unsafeAtomicAdd(&denom[dst], x);   // native global_atomic_add_f32, no return
}

// ---------------------------------------------------------------------------
// Kernel 4: wave per edge; coalesced gather of Wh[src], scatter-add to out[dst]
// ---------------------------------------------------------------------------
__global__ void gat_edge_scatter(const long long* __restrict__ adj,
                                 const float* __restrict__ Wh,
                                 const float* __restrict__ x_exp,
                                 const float* __restrict__ denom,
                                 float* __restrict__ out, int E) {
    const int lane = threadIdx.x & 31;
    const int wave = (int)(((size_t)blockIdx.x * blockDim.x + threadIdx.x) >> 5);
    if (wave >= E) return;
    long long dst = adj[2 * (size_t)wave + 0];
    long long src = adj[2 * (size_t)wave + 1];
    const float att = x_exp[wave] / (denom[dst] + DENOM_EPS);
    const float* wsrc = Wh + (size_t)src * OUT_F;
    float* od = out + (size_t)dst * OUT_F;
    unsafeAtomicAdd(&od[lane],      wsrc[lane]      * att);
    unsafeAtomicAdd(&od[lane + 32], wsrc[lane + 32] * att);
}

// ---------------------------------------------------------------------------
// Kernel 5: ELU in place on out
// ---------------------------------------------------------------------------
__global__ void gat_elu(float* __restrict__ out, int n) {
    int i = blockIdx.x * blockDim.x + threadIdx.x;
    if (i >= n) return;
    float v = out[i];
    out[i] = v > 0.0f ? v : (__expf(v) - 1.0f);
}

// ---------------------------------------------------------------------------
extern "C" void kernel_launch(void* const* d_in, const int* in_sizes, int n_in,
                              void* d_out, int out_size, void* d_ws, size_t ws_size,
                              hipStream_t stream) {
    const float* h = (const float*)d_in[0];       // [N, 128]
    const float* W = (const float*)d_in[1];       // [128, 64]
    const float* a = (const float*)d_in[2];       // [128]
    const long long* adj = (const long long*)d_in[3]; // [E, 2] int64

    const int N = in_sizes[0] / IN_F;
    const int E = in_sizes[3] / 2;

    // Workspace layout (floats)
    float* ws = (float*)d_ws;
    float* Wh = ws;                           // N*64
    float* s_arr = Wh + (size_t)N * OUT_F;    // N
    float* t_arr = s_arr + N;                 // N
    float* denom = t_arr + N;                 // N
    float* x_exp = denom + N;                 // E

    float* out = (float*)d_out;               // N*64, accumulated via atomics

    const int BLK = 256;

    // 0) zero out + denom
    {
        int n_out = N * OUT_F;
        int grid = (n_out + BLK - 1) / BLK;
        gat_zero_init<<<grid, BLK, 0, stream>>>(out, denom, n_out, N);
    }

    // 1) Wh = h @ W  (WMMA f32 16x16x4)
    {
        int mTiles = (N + 15) / 16;
        int nTiles = OUT_F / 16;
        int waves = mTiles * nTiles;
        int wavesPerBlock = BLK / 32;
        int grid = (waves + wavesPerBlock - 1) / wavesPerBlock;
        gat_wh_gemm<<<grid, BLK, 0, stream>>>(h, W, Wh, N);
    }

    // 2) s, t
    {
        int grid = (N + BLK - 1) / BLK;
        gat_st<<<grid, BLK, 0, stream>>>(Wh, a, s_arr, t_arr, N);
    }

    // 3) edge exp + denom segment sum
    {
        int grid = (E + BLK - 1) / BLK;
        gat_edge_exp<<<grid, BLK, 0, stream>>>(adj, s_arr, t_arr, x_exp, denom, E);
    }

    // 4) scatter messages (wave per edge)
    {
        int wavesPerBlock = BLK / 32;
        int grid = (E + wavesPerBlock - 1) / wavesPerBlock;
        gat_edge_scatter<<<grid, BLK, 0, stream>>>(adj, Wh, x_exp, denom, out, E);
    }

    // 5) ELU
    {
        int n_out = N * OUT_F;
        int grid = (n_out + BLK - 1) / BLK;
        gat_elu<<<grid, BLK, 0, stream>>>(out, n_out);
    }
}